// VectorQuantizer_1271310320158
// MI455X (gfx1250) — compile-verified
//
#include <hip/hip_runtime.h>

typedef __attribute__((ext_vector_type(2))) float v2f;
typedef __attribute__((ext_vector_type(8))) float v8f;

#define N_ROWS   131072
#define DIM      64
#define K_CODES  512
#define CHUNK    64        // codes staged in LDS at a time
#define N_CHUNKS (K_CODES / CHUNK)                      // 8
#define LDS_STRIDE 68      // dwords per code row in LDS; 68 % 64 == 4 -> conflict-free b64 reads
#define N_WAVES_PER_BLOCK 8
#define N_BLOCKS (N_ROWS / (16 * N_WAVES_PER_BLOCK))   // 1024
#define N_PARTIALS (N_BLOCKS * N_WAVES_PER_BLOCK)       // 8192
#define BETA 0.25f

// ---------------- prolog: per-code squared norms -> ws[0..511] ----------------
__global__ __launch_bounds__(256) void vq_norms_kernel(const float* __restrict__ cb,
                                                       float* __restrict__ norms) {
  int n = blockIdx.x * 256 + threadIdx.x;
  if (n < K_CODES) {
    float s = 0.f;
    #pragma unroll 8
    for (int k = 0; k < DIM; ++k) { float v = cb[(size_t)n * DIM + k]; s += v * v; }
    norms[n] = s;
  }
}

// ---------------- main: WMMA distance + argmin + gather + partial loss ----------------
__global__ __launch_bounds__(256) void vq_main_kernel(const float* __restrict__ x,
                                                      const float* __restrict__ cb,
                                                      const float* __restrict__ norms,
                                                      float* __restrict__ out,
                                                      float* __restrict__ partials) {
  // double-buffered codebook chunks, rows padded to 68 dwords (272 B = 17x16, b128-aligned)
  __shared__ __align__(16) float scb[2][CHUNK * LDS_STRIDE];   // ~35 KB

  const int tid  = threadIdx.x;
  const int lane = tid & 31;
  const int wave = tid >> 5;
  const int gwave    = blockIdx.x * N_WAVES_PER_BLOCK + wave;
  const int row_base = gwave * 16;

  const int m    = lane & 15;   // row-in-tile (A) / code-in-tile (B,C)
  const int half = lane >> 4;   // selects K pair per ISA fragment layout

  // async-copy one 64x64 f32 codebook chunk into LDS buffer `buf` (no VGPR transit)
  auto issue_chunk = [&](int c, int buf) {
    #pragma unroll
    for (int j = 0; j < 4; ++j) {
      int f  = j * 256 + tid;          // float4 index 0..1023
      int n  = f >> 4;                 // 0..63
      int k4 = (f & 15) << 2;          // 0,4,...,60
      unsigned long long ga = (unsigned long long)(uintptr_t)
          (cb + ((size_t)(c * CHUNK + n)) * DIM + k4);
      unsigned int la = (unsigned int)(uintptr_t)&scb[buf][n * LDS_STRIDE + k4];
      asm volatile("global_load_async_to_lds_b128 %0, %1, off"
                   :: "v"(la), "v"(ga) : "memory");
    }
  };

  // ---- A fragments: x tile 16x64 as 16 slices of 16x4 f32 ----
  // lane layout (ISA 7.12.2): v.x = x[row, 4s + 2*half], v.y = x[row, 4s + 2*half + 1]
  v2f a[16];
  {
    const float* xr = x + (size_t)(row_base + m) * DIM + 2 * half;
    #pragma unroll
    for (int s = 0; s < 16; ++s) { a[s].x = xr[4 * s]; a[s].y = xr[4 * s + 1]; }
  }

  float minv[8];
  int   mini[8];
  #pragma unroll
  for (int r = 0; r < 8; ++r) { minv[r] = 3.0e38f; mini[r] = 0; }

  issue_chunk(0, 0);

  for (int c = 0; c < N_CHUNKS; ++c) {
    // my async writes for chunk c have landed; barrier makes everyone's visible,
    // and guarantees all reads of the buffer we are about to refill are done.
    asm volatile("s_wait_asynccnt 0x0" ::: "memory");
    __syncthreads();
    if (c + 1 < N_CHUNKS) issue_chunk(c + 1, (c + 1) & 1);

    const float* sbase = &scb[c & 1][0];
    #pragma unroll
    for (int t = 0; t < CHUNK / 16; ++t) {
      v8f acc = {};
      // B fragment (4x16, codes as columns): b.x = cb[n, 4s + 2*half], b.y = next k
      const float* bp = sbase + (t * 16 + m) * LDS_STRIDE + 2 * half;
      #pragma unroll
      for (int s = 0; s < 16; ++s) {
        v2f b; b.x = bp[4 * s]; b.y = bp[4 * s + 1];
        acc = __builtin_amdgcn_wmma_f32_16x16x4_f32(
            /*neg_a=*/false, a[s], /*neg_b=*/false, b,
            /*c_mod=*/(short)0, acc, /*reuse_a=*/false, /*reuse_b=*/false);
      }
      // C layout: VGPR r -> row (r + 8*half), code n = base + (lane&15)
      int   n_glob = c * CHUNK + t * 16 + m;
      float nrm    = norms[n_glob];
      #pragma unroll
      for (int r = 0; r < 8; ++r) {
        float score = nrm - 2.0f * acc[r];          // ||e||^2 - 2 x.e  (||x||^2 const per row)
        if (score < minv[r]) { minv[r] = score; mini[r] = n_glob; }  // strict < keeps first idx
      }
    }
  }

  // ---- argmin across the 16 codes held per lane-group (butterfly within each half) ----
  #pragma unroll
  for (int r = 0; r < 8; ++r) {
    #pragma unroll
    for (int msk = 8; msk >= 1; msk >>= 1) {
      float ov = __shfl_xor(minv[r], msk, 32);
      int   oi = __shfl_xor(mini[r], msk, 32);
      if (ov < minv[r] || (ov == minv[r] && oi < mini[r])) { minv[r] = ov; mini[r] = oi; }
    }
  }

  // ---- gather codebook rows, write straight-through output, accumulate loss ----
  float ls = 0.f;
  #pragma unroll
  for (int r = 0; r < 8; ++r) {
    int ia = __shfl(mini[r], 0, 32);    // row row_base + r
    int ib = __shfl(mini[r], 16, 32);   // row row_base + r + 8
    {
      size_t o = (size_t)(row_base + r) * DIM + 2 * lane;
      float2 q  = *(const float2*)(cb + (size_t)ia * DIM + 2 * lane);
      float2 xv = *(const float2*)(x + o);
      *(float2*)(out + o) = q;          // quantized_st == quantized numerically
      float dx = q.x - xv.x, dy = q.y - xv.y;
      ls += dx * dx + dy * dy;
    }
    {
      size_t o = (size_t)(row_base + r + 8) * DIM + 2 * lane;
      float2 q  = *(const float2*)(cb + (size_t)ib * DIM + 2 * lane);
      float2 xv = *(const float2*)(x + o);
      *(float2*)(out + o) = q;
      float dx = q.x - xv.x, dy = q.y - xv.y;
      ls += dx * dx + dy * dy;
    }
  }
  #pragma unroll
  for (int msk = 16; msk >= 1; msk >>= 1) ls += __shfl_xor(ls, msk, 32);
  if (lane == 0) partials[gwave] = ls;
}

// ---------------- epilog: deterministic loss reduction ----------------
__global__ __launch_bounds__(256) void vq_reduce_kernel(const float* __restrict__ partials,
                                                        float* __restrict__ loss_out) {
  __shared__ float sd[256];
  int t = threadIdx.x;
  float s = 0.f;
  for (int i = t; i < N_PARTIALS; i += 256) s += partials[i];
  sd[t] = s;
  __syncthreads();
  for (int w = 128; w > 0; w >>= 1) { if (t < w) sd[t] += sd[t + w]; __syncthreads(); }
  if (t == 0) *loss_out = sd[0] * (BETA / (float)(N_ROWS * DIM));
}

extern "C" void kernel_launch(void* const* d_in, const int* in_sizes, int n_in,
                              void* d_out, int out_size, void* d_ws, size_t ws_size,
                              hipStream_t stream) {
  const float* x  = (const float*)d_in[0];
  const float* cb = (const float*)d_in[1];
  float* out      = (float*)d_out;

  float* norms    = (float*)d_ws;            // 512 floats
  float* partials = norms + K_CODES;         // 8192 floats

  vq_norms_kernel<<<(K_CODES + 255) / 256, 256, 0, stream>>>(cb, norms);
  vq_main_kernel<<<N_BLOCKS, 256, 0, stream>>>(x, cb, norms, out, partials);
  vq_reduce_kernel<<<1, 256, 0, stream>>>(partials, out + (size_t)N_ROWS * DIM);
}